// TensorProductUniform1d_59356448030867
// MI455X (gfx1250) — compile-verified
//
#include <hip/hip_runtime.h>

// out[b, k*128 + e] = (-1)^k * sum_i in0[b, i*128+e] * in1[b, ((k-i) mod 8)*128 + e]
// Signed 8-point circular convolution per (b, e). Memory-bound: 768 MB @ 23.3 TB/s => ~33 us floor.
// Strategy: single-pass streaming, float4 (b128) coalesced non-temporal loads/stores,
// fully unrolled FMA chain (64 vector FMAs / thread over 4 lanes each).

typedef float v4f __attribute__((ext_vector_type(4)));

__global__ __launch_bounds__(256) void conv8_seg_kernel(
    const float* __restrict__ in0,
    const float* __restrict__ in1,
    float* __restrict__ out,
    int n_threads)  // = B * (E/4)
{
    const int tid = blockIdx.x * blockDim.x + threadIdx.x;
    if (tid >= n_threads) return;

    // E = 128 floats per segment, S = 8 segments, row = 1024 floats.
    // b = tid / 32, e4 = tid % 32 ; base float offset = b*1024 + e4*4
    const int base = ((tid >> 5) << 10) + ((tid & 31) << 2);

    const v4f* __restrict__ p0 = (const v4f*)(in0 + base);
    const v4f* __restrict__ p1 = (const v4f*)(in1 + base);
    v4f*       __restrict__ po = (v4f*)(out + base);

    // Segment stride = 128 floats = 32 v4f.
    v4f u[8], v[8];
#pragma unroll
    for (int s = 0; s < 8; ++s) {
        u[s] = __builtin_nontemporal_load(p0 + s * 32);
        v[s] = __builtin_nontemporal_load(p1 + s * 32);
    }

#pragma unroll
    for (int k = 0; k < 8; ++k) {
        v4f acc = u[0] * v[k];            // i = 0, j = k
#pragma unroll
        for (int i = 1; i < 8; ++i) {
            const int j = (k - i) & 7;    // compile-time constant after unroll
            acc += u[i] * v[j];           // contracts to v_fma_f32 / v_pk_fma_f32
        }
        if (k & 1) acc = -acc;            // (-1)^k
        __builtin_nontemporal_store(acc, po + k * 32);
    }
}

extern "C" void kernel_launch(void* const* d_in, const int* in_sizes, int n_in,
                              void* d_out, int out_size, void* d_ws, size_t ws_size,
                              hipStream_t stream) {
    const float* in0 = (const float*)d_in[0];
    const float* in1 = (const float*)d_in[1];
    float* out = (float*)d_out;

    // in_sizes[0] = B * S * E = B * 1024 floats; each thread covers 32 floats of each input.
    const long long total = (long long)in_sizes[0];
    const int n_threads = (int)(total / 32);

    const int block = 256;  // 8 wave32 per workgroup
    const int grid = (n_threads + block - 1) / block;
    conv8_seg_kernel<<<grid, block, 0, stream>>>(in0, in1, out, n_threads);
}